// Memory_72249939853591
// MI455X (gfx1250) — compile-verified
//
#include <hip/hip_runtime.h>
#include <math.h>

// NTM memory write head for MI455X (gfx1250, wave32).
// N = 131072 rows, M = 256 cols, fp32 end-to-end.
// Pass 1 (matvec + row norms) uses V_WMMA_F32_16X16X4_F32.

#define NROWS 131072
#define MCOLS 256
#define EPSV  1e-8f

typedef __attribute__((ext_vector_type(2))) float v2f;
typedef __attribute__((ext_vector_type(8))) float v8f;

// -------- ws (float) layout --------
// [0..255]   kt = tanh(k)
// [256]      k_norm (clamped to eps)
// [257]      g = sigmoid(g)
// [258..260] s softmax
// [261]      gamma = relu+1
// [262]      1 / sum(exp(score))
// [263]      1 / sum(ws^gamma)
// [320..1343]  pass-1 block partials (1024)
// [2048..2559] pass-2 block partials (512)

__global__ void k0_scalars(const float* __restrict__ k, const float* __restrict__ g,
                           const float* __restrict__ s, const float* __restrict__ gamma,
                           float* __restrict__ ws) {
    __shared__ float red[256];
    int t = threadIdx.x;
    float kt = tanhf(k[t]);
    ws[t] = kt;
    red[t] = kt * kt;
    __syncthreads();
    for (int off = 128; off > 0; off >>= 1) {
        if (t < off) red[t] += red[t + off];
        __syncthreads();
    }
    if (t == 0) {
        ws[256] = fmaxf(sqrtf(red[0]), EPSV);
        ws[257] = 1.0f / (1.0f + expf(-g[0]));
        float s0 = s[0], s1 = s[1], s2 = s[2];
        float mx = fmaxf(s0, fmaxf(s1, s2));
        float e0 = expf(s0 - mx), e1 = expf(s1 - mx), e2 = expf(s2 - mx);
        float inv = 1.0f / (e0 + e1 + e2);
        ws[258] = e0 * inv; ws[259] = e1 * inv; ws[260] = e2 * inv;
        ws[261] = fmaxf(gamma[0], 0.0f) + 1.0f;
    }
}

// Pass 1: per wave, 16 rows. dots via WMMA f32 16x16x4 (B = broadcast k-chunk),
// row sum-of-squares via VALU on the same A registers + shfl_xor(16).
__global__ void __launch_bounds__(256)
k1_scores(const float* __restrict__ mem, const float* __restrict__ ws,
          float* __restrict__ enum_out, float* __restrict__ partials) {
    __shared__ float skt[256];
    __shared__ float red[256];
    const int t = threadIdx.x;
    skt[t] = ws[t];                       // stage tanh(k) in LDS (broadcast reads)
    __syncthreads();

    const int wave = t >> 5;
    const int lane = t & 31;
    const int rowbase = blockIdx.x * 128 + wave * 16;
    const int r15  = lane & 15;
    const int koff = (lane >= 16) ? 2 : 0;   // ISA A/B layout: hi lanes hold K=2,3

    const float* rp = mem + (size_t)(rowbase + r15) * MCOLS + koff;
    const float* kp = skt + koff;

    v8f acc = {0.f, 0.f, 0.f, 0.f, 0.f, 0.f, 0.f, 0.f};
    float ss = 0.0f;

#pragma unroll 4
    for (int c = 0; c < MCOLS / 4; ++c) {
        v2f a = *(const v2f*)(rp + 4 * c);    // A: 16x4 fp32 tile (this wave's rows)
        v2f b = *(const v2f*)(kp + 4 * c);    // B: k-chunk broadcast to all 16 cols
        ss = fmaf(a[0], a[0], ss);
        ss = fmaf(a[1], a[1], ss);
        // D = A x B + C   (dots accumulate in every column; we read col of choice)
        acc = __builtin_amdgcn_wmma_f32_16x16x4_f32(false, a, false, b,
                                                    (short)0, acc, false, false);
    }

    // full row sumsq: combine the two half-lanes holding the same row
    ss += __shfl_xor(ss, 16, 32);            // lane l now holds sumsq[row (l&15)]

    // Extract dots: lanes 0-15 hold rows 0-7 in acc[0..7]; lanes 16-31 rows 8-15.
    const int j = lane & 7;
    float d = acc[0];
    if (j == 1) d = acc[1];
    if (j == 2) d = acc[2];
    if (j == 3) d = acc[3];
    if (j == 4) d = acc[4];
    if (j == 5) d = acc[5];
    if (j == 6) d = acc[6];
    if (j == 7) d = acc[7];
    const bool active = (lane & 8) == 0;          // lanes 0-7 and 16-23 write
    const int  r = (lane & 7) + ((lane >> 4) << 3); // row 0..15 within tile
    const float ssr = __shfl(ss, r, 32);          // sumsq for that row
    const float kn  = ws[256];

    float en = 0.0f;
    if (active) {
        float mn    = fmaxf(sqrtf(ssr), EPSV);
        float score = d / (mn * kn);              // cosine in [-1,1] -> safe exp
        en = expf(score);
        enum_out[rowbase + r] = en;
    }

    red[t] = en;
    __syncthreads();
    for (int off = 128; off > 0; off >>= 1) {
        if (t < off) red[t] += red[t + off];
        __syncthreads();
    }
    if (t == 0) partials[blockIdx.x] = red[0];
}

// Deterministic partial-sum reduce; stores reciprocal of the total.
__global__ void k_reduce_inv(const float* __restrict__ partials, int n,
                             float* __restrict__ out) {
    __shared__ float red[1024];
    int t = threadIdx.x;
    float s = 0.0f;
    for (int i = t; i < n; i += 1024) s += partials[i];
    red[t] = s;
    __syncthreads();
    for (int off = 512; off > 0; off >>= 1) {
        if (t < off) red[t] += red[t + off];
        __syncthreads();
    }
    if (t == 0) out[0] = 1.0f / red[0];
}

// Pass 2: wc -> interpolate -> circular 3-tap shift -> pow(gamma); partial sums.
__global__ void __launch_bounds__(256)
k2_shift_pow(const float* __restrict__ enum_in, const float* __restrict__ prev_w,
             const float* __restrict__ ws, float* __restrict__ wpow_out,
             float* __restrict__ partials) {
    __shared__ float red[256];
    const int t = threadIdx.x;
    const int n = blockIdx.x * 256 + t;
    const float inv = ws[262];
    const float gv  = ws[257];
    const float s0 = ws[258], s1 = ws[259], s2 = ws[260], gam = ws[261];

    auto wg = [&](int i) {
        return gv * (enum_in[i] * inv) + (1.0f - gv) * prev_w[i];
    };
    const int nm1 = (n + NROWS - 1) & (NROWS - 1);
    const int np1 = (n + 1) & (NROWS - 1);
    float wsv = s0 * wg(nm1) + s1 * wg(n) + s2 * wg(np1);   // ws > 0 always
    float wp  = powf(wsv, gam);
    wpow_out[n] = wp;

    red[t] = wp;
    __syncthreads();
    for (int off = 128; off > 0; off >>= 1) {
        if (t < off) red[t] += red[t + off];
        __syncthreads();
    }
    if (t == 0) partials[blockIdx.x] = red[0];
}

__global__ void k2b_normalize(float* __restrict__ w, const float* __restrict__ inv) {
    int n = blockIdx.x * 256 + threadIdx.x;
    w[n] *= inv[0];
}

// Pass 3: new_memory = memory*(1 - w*e) + w*a, float4 streaming (4 rows/block).
__global__ void __launch_bounds__(256)
k3_update(const float* __restrict__ mem, const float* __restrict__ e,
          const float* __restrict__ a, const float* __restrict__ w,
          float* __restrict__ out) {
    const int t   = threadIdx.x;
    const int row = blockIdx.x * 4 + (t >> 6);
    const int c   = t & 63;
    const float wv = w[row];
    float4 mv = ((const float4*)(mem + (size_t)row * MCOLS))[c];
    float4 ev = ((const float4*)e)[c];
    float4 av = ((const float4*)a)[c];
    float4 o;
    o.x = mv.x * (1.0f - wv * ev.x) + wv * av.x;
    o.y = mv.y * (1.0f - wv * ev.y) + wv * av.y;
    o.z = mv.z * (1.0f - wv * ev.z) + wv * av.z;
    o.w = mv.w * (1.0f - wv * ev.w) + wv * av.w;
    ((float4*)(out + (size_t)row * MCOLS))[c] = o;
}

extern "C" void kernel_launch(void* const* d_in, const int* in_sizes, int n_in,
                              void* d_out, int out_size, void* d_ws, size_t ws_size,
                              hipStream_t stream) {
    const float* mem    = (const float*)d_in[0];
    const float* k      = (const float*)d_in[1];
    // d_in[2] = beta: unused by the reference computation (faithful)
    const float* g      = (const float*)d_in[3];
    const float* s      = (const float*)d_in[4];
    const float* gamma  = (const float*)d_in[5];
    const float* prev_w = (const float*)d_in[6];
    const float* e      = (const float*)d_in[7];
    const float* a      = (const float*)d_in[8];

    float* out = (float*)d_out;
    float* ws  = (float*)d_ws;

    float* enum_buf = out;                            // N floats; overwritten by k3
    float* wbuf     = out + (size_t)NROWS * MCOLS;    // final w output slot
    float* part1    = ws + 320;                       // 1024 partials
    float* part2    = ws + 2048;                      // 512 partials

    k0_scalars  <<<1,     256, 0, stream>>>(k, g, s, gamma, ws);
    k1_scores   <<<NROWS / 128, 256, 0, stream>>>(mem, ws, enum_buf, part1);
    k_reduce_inv<<<1,    1024, 0, stream>>>(part1, NROWS / 128, ws + 262);
    k2_shift_pow<<<NROWS / 256, 256, 0, stream>>>(enum_buf, prev_w, ws, wbuf, part2);
    k_reduce_inv<<<1,    1024, 0, stream>>>(part2, NROWS / 256, ws + 263);
    k2b_normalize<<<NROWS / 256, 256, 0, stream>>>(wbuf, ws + 263);
    k3_update   <<<NROWS / 4,   256, 0, stream>>>(mem, e, a, wbuf, out);
}